// GraphSAGELayer_38354057954017
// MI455X (gfx1250) — compile-verified
//
#include <hip/hip_runtime.h>

// GraphSAGE layer for MI455X (gfx1250, wave32):
//   agg = scatter-add(w_edge * x[src]) ; y = [x|agg] @ W^T (fp32 WMMA) ; BN(y)
//
// ws layout (floats): agg[N*F_IN] | stats[512] (sum,sumsq,scale,shift x128)
// y lives directly in d_out; BN applied in place.

#define F_IN   128
#define F_OUT  128
#define K_DIM  256
#define BN_EPS 1e-5f

#define M_BLK      48            // rows of y per block (3 subtiles of 16)
#define M_SUB      (M_BLK / 16)
#define LDS_STRIDE 260           // floats; 260 % 64 == 4 -> conflict-free b64 reads

typedef __attribute__((ext_vector_type(2))) float v2f;
typedef __attribute__((ext_vector_type(8))) float v8f;

__device__ __forceinline__ void atomic_fadd(float* p, float v) {
  unsafeAtomicAdd(p, v);   // native global_atomic_add_f32 (no CAS loop)
}

// ---------------------------------------------------------------- K1: zero
__global__ void sage_zero_kernel(float* __restrict__ p, int n) {
  int i = blockIdx.x * blockDim.x + threadIdx.x;
  if (i < n) p[i] = 0.0f;
}

// ------------------------------------------------- K2: edge scatter (SpMM)
// One wave32 per edge: lane L moves 4 consecutive feats (float4 = 512B/row).
__global__ void __launch_bounds__(256)
sage_spmm_kernel(const float* __restrict__ x, const int* __restrict__ src,
                 const int* __restrict__ dst, const float* __restrict__ w_edge,
                 float* __restrict__ agg, int E) {
  int gid  = blockIdx.x * blockDim.x + threadIdx.x;
  int e    = gid >> 5;
  int lane = gid & 31;
  if (e >= E) return;
  int s = src[e];
  int d = dst[e];
  float w = w_edge[e];
  const float4 v = ((const float4*)(x + (size_t)s * F_IN))[lane];
  float* a = agg + (size_t)d * F_IN + lane * 4;
  atomic_fadd(a + 0, w * v.x);
  atomic_fadd(a + 1, w * v.y);
  atomic_fadd(a + 2, w * v.z);
  atomic_fadd(a + 3, w * v.w);
}

// ------------------------------------- K3: fp32 WMMA GEMM + fused BN stats
// Block = 48-row strip of y staged in LDS; wave w owns column tile [16w,16w+16).
// Each wave: 3 M-subtiles, one B fragment per k-step reused across subtiles.
__global__ void __launch_bounds__(256)
sage_gemm_kernel(const float* __restrict__ x, const float* __restrict__ agg,
                 const float* __restrict__ Wmat, float* __restrict__ y,
                 float* __restrict__ stats, int N) {
  __shared__ float hA[M_BLK * LDS_STRIDE];   // 48 x 260 floats = 49.9 KB

  const int tid     = threadIdx.x;
  const int rowBase = blockIdx.x * M_BLK;

  // Cooperative fill: 48 rows x 64 float4 (quads 0..31 from x, 32..63 from agg).
  // Consecutive threads write consecutive float4 -> conflict-free ds_store_b128.
  for (int idx = tid; idx < M_BLK * 64; idx += 256) {
    const int r  = idx >> 6;
    const int q  = idx & 63;
    const int gr = min(rowBase + r, N - 1);          // clamp (guarded at store)
    const float4 v = (q < 32)
        ? ((const float4*)(x   + (size_t)gr * F_IN))[q]
        : ((const float4*)(agg + (size_t)gr * F_IN))[q - 32];
    *(float4*)&hA[r * LDS_STRIDE + 4 * q] = v;
  }
  __syncthreads();

  const int wave    = tid >> 5;        // 0..7 -> column tile
  const int lane    = tid & 31;
  const int half    = lane >> 4;       // K sub-slot for A/B layout
  const int j       = lane & 15;       // M index (A) / N index (B)
  const int colBase = wave * 16;

  // y[m,n] = sum_k h[m,k] * W[n,k]  ->  B[k][n] = W[n][k] (row n, contiguous k)
  const float* __restrict__ brow = Wmat + (size_t)(colBase + j) * K_DIM;

  v8f acc[M_SUB];
  #pragma unroll
  for (int s = 0; s < M_SUB; ++s) acc[s] = (v8f){};

  // A 16x4 f32 layout: lane = m + 16*half, v2f = {K(k0+2h), K(k0+2h+1)}
  // B 4x16 f32 layout: lane = n + 16*half, same K pairing.
  for (int k0 = 0; k0 < K_DIM; k0 += 4) {
    const int k = k0 + 2 * half;
    v2f b; b.x = brow[k]; b.y = brow[k + 1];
    #pragma unroll
    for (int s = 0; s < M_SUB; ++s) {
      const float* ap = &hA[(s * 16 + j) * LDS_STRIDE + k];
      v2f a; a.x = ap[0]; a.y = ap[1];
      acc[s] = __builtin_amdgcn_wmma_f32_16x16x4_f32(false, a, false, b,
                                                     (short)0, acc[s],
                                                     false, false);
    }
  }

  // C/D layout: VGPR r -> row rowBase + s*16 + r + 8*half, col = colBase + j.
  // All elements of this lane share one column -> fuse BN column sums.
  const int col = colBase + j;
  float ssum = 0.0f, ssq = 0.0f;
  #pragma unroll
  for (int s = 0; s < M_SUB; ++s) {
    #pragma unroll
    for (int r = 0; r < 8; ++r) {
      const int row = rowBase + s * 16 + r + 8 * half;
      const float v = acc[s][r];
      if (row < N) {
        y[(size_t)row * F_OUT + col] = v;
        ssum += v;
        ssq  += v * v;
      }
    }
  }
  atomic_fadd(&stats[col], ssum);          // sum_y
  atomic_fadd(&stats[F_OUT + col], ssq);   // sum_y^2
}

// ------------------------------------ K4: fold stats -> per-col scale/shift
__global__ void sage_bnstats_kernel(float* __restrict__ stats,
                                    const float* __restrict__ gamma,
                                    const float* __restrict__ beta, float invN) {
  const int c = threadIdx.x;   // 128 threads
  const float mean = stats[c] * invN;
  const float var  = stats[F_OUT + c] * invN - mean * mean;  // biased var
  const float sc   = gamma[c] * rsqrtf(var + BN_EPS);
  stats[2 * F_OUT + c] = sc;
  stats[3 * F_OUT + c] = beta[c] - mean * sc;
}

// ----------------------------------------- K5: apply BN in place on d_out
__global__ void __launch_bounds__(256)
sage_bnapply_kernel(float4* __restrict__ y4, const float* __restrict__ stats,
                    int total4) {
  int i = blockIdx.x * blockDim.x + threadIdx.x;
  if (i >= total4) return;
  const int col = (i * 4) & (F_OUT - 1);
  const float* __restrict__ sc = stats + 2 * F_OUT;
  const float* __restrict__ sh = stats + 3 * F_OUT;
  float4 v = y4[i];
  v.x = v.x * sc[col + 0] + sh[col + 0];
  v.y = v.y * sc[col + 1] + sh[col + 1];
  v.z = v.z * sc[col + 2] + sh[col + 2];
  v.w = v.w * sc[col + 3] + sh[col + 3];
  y4[i] = v;
}

extern "C" void kernel_launch(void* const* d_in, const int* in_sizes, int n_in,
                              void* d_out, int out_size, void* d_ws, size_t ws_size,
                              hipStream_t stream) {
  const float* x      = (const float*)d_in[0];
  const int*   src    = (const int*)  d_in[1];
  const int*   dst    = (const int*)  d_in[2];
  const float* w_edge = (const float*)d_in[3];
  const float* Wmat   = (const float*)d_in[4];
  const float* gamma  = (const float*)d_in[5];
  const float* beta   = (const float*)d_in[6];

  const int N = in_sizes[0] / F_IN;
  const int E = in_sizes[1];

  float* agg   = (float*)d_ws;                 // N*F_IN floats
  float* stats = agg + (size_t)N * F_IN;       // 512 floats: sum|sumsq|scale|shift
  float* y     = (float*)d_out;                // GEMM output lives in d_out

  // K1: zero agg + stats (contiguous)
  {
    int zn = N * F_IN + 4 * F_OUT;
    sage_zero_kernel<<<(zn + 255) / 256, 256, 0, stream>>>(agg, zn);
  }
  // K2: edge scatter-add (one wave per edge)
  {
    long long threads = (long long)E * 32;
    int blocks = (int)((threads + 255) / 256);
    sage_spmm_kernel<<<blocks, 256, 0, stream>>>(x, src, dst, w_edge, agg, E);
  }
  // K3: WMMA GEMM (LDS-staged A) + fused column stats
  {
    int mBlocks = (N + M_BLK - 1) / M_BLK;
    sage_gemm_kernel<<<mBlocks, 256, 0, stream>>>(x, agg, Wmat, y, stats, N);
  }
  // K4: fold stats into scale/shift
  sage_bnstats_kernel<<<1, F_OUT, 0, stream>>>(stats, gamma, beta, 1.0f / (float)N);
  // K5: normalize in place
  {
    int total4 = (N * F_OUT) / 4;
    sage_bnapply_kernel<<<(total4 + 255) / 256, 256, 0, stream>>>((float4*)y, stats, total4);
  }
}